// SNU_Net_42631845380649
// MI455X (gfx1250) — compile-verified
//
#include <hip/hip_runtime.h>
#include <stdint.h>

typedef float v2f __attribute__((ext_vector_type(2)));
typedef float v8f __attribute__((ext_vector_type(8)));

// ---- problem constants (from reference) ----
#define T_STEPS 1000
#define BATCH   64
#define K_DIM   1024
#define N_DIM   2048
#define M_DIM   (T_STEPS * BATCH)   // 64000
#define THRESH  1.0f

// ---- GEMM tiling ----
#define BM 128
#define BN 256
#define BK 16
#define NKT (K_DIM / BK)            // 64 K-tiles
#define ASTRIDE 20                  // BK+4: b128-aligned LDS rows, conflict-free frag reads
#define BSTRIDE 260                 // BN+4: b128-aligned LDS rows
#define A_TILE (BM * ASTRIDE)       // 2560 floats
#define B_TILE (BK * BSTRIDE)       // 4160 floats

// ---- CDNA5 async global->LDS copy (ASYNCcnt-tracked), inline asm ----
__device__ __forceinline__ void async_b128(uint32_t lds_off, const float* g) {
    asm volatile("global_load_async_to_lds_b128 %0, %1, off"
                 :: "v"(lds_off), "v"(g) : "memory");
}
__device__ __forceinline__ void wait_async_le6() {   // keep newest 6 in flight
    asm volatile("s_wait_asynccnt 0x6" ::: "memory");
}
__device__ __forceinline__ void wait_async_0() {
    asm volatile("s_wait_asynccnt 0x0" ::: "memory");
}

// Issue one K-tile's staging: A 128x16 (2 b128/thread) + B 16x256 (4 b128/thread).
__device__ __forceinline__ void issue_tile(const float* __restrict__ X,
                                           const float* __restrict__ W,
                                           int m0, int n0, int kt,
                                           uint32_t aBase, uint32_t bBase, int tid)
{
    {   // A tile: row-major BM x BK, 16 cols = 4 float4 per row
        const int row = tid >> 2;
        const int cv  = (tid & 3) * 4;
        const float* g = X + (size_t)(m0 + row) * K_DIM + kt + cv;
        async_b128(aBase + (uint32_t)(row * ASTRIDE + cv) * 4u, g);
        async_b128(aBase + (uint32_t)((row + 64) * ASTRIDE + cv) * 4u,
                   g + (size_t)64 * K_DIM);
    }
    {   // B tile: row-major BK x BN, 256 cols = 64 float4 per row
        const int row = tid >> 6;              // 0..3
        const int cv  = (tid & 63) * 4;
        const float* g = W + (size_t)(kt + row) * N_DIM + n0 + cv;
        uint32_t l = bBase + (uint32_t)(row * BSTRIDE + cv) * 4u;
#pragma unroll
        for (int r = 0; r < 4; ++r) {          // rows {row, row+4, row+8, row+12}
            async_b128(l, g);
            g += (size_t)4 * N_DIM;
            l += 4u * BSTRIDE * 4u;
        }
    }
}

// Phase 1: Z = X @ W via V_WMMA_F32_16X16X4_F32.
// 256 threads = 8 waves (2x4); wave tile 64x64 = 4x4 accumulators (128 VGPRs).
// Double-buffered async global->LDS staging.
__launch_bounds__(256)
__global__ void snu_gemm_f32(const float* __restrict__ X,
                             const float* __restrict__ W,
                             float* __restrict__ Z)
{
    __shared__ float As[2][A_TILE];
    __shared__ float Bs[2][B_TILE];

    const int tid   = threadIdx.x;
    const int lid   = tid & 31;
    const int w     = tid >> 5;
    const int wr    = w >> 2;        // 0..1 along M
    const int wc    = w & 3;         // 0..3 along N
    const int lrow  = lid & 15;
    const int khalf = lid >> 4;

    const int m0 = blockIdx.y * BM;
    const int n0 = blockIdx.x * BN;

    const uint32_t aB[2] = { (uint32_t)(uintptr_t)&As[0][0],
                             (uint32_t)(uintptr_t)&As[1][0] };
    const uint32_t bB[2] = { (uint32_t)(uintptr_t)&Bs[0][0],
                             (uint32_t)(uintptr_t)&Bs[1][0] };

    const v8f vzero = {0.f, 0.f, 0.f, 0.f, 0.f, 0.f, 0.f, 0.f};
    v8f acc[4][4];
#pragma unroll
    for (int i = 0; i < 4; ++i)
#pragma unroll
        for (int j = 0; j < 4; ++j) acc[i][j] = vzero;

    issue_tile(X, W, m0, n0, 0, aB[0], bB[0], tid);

    for (int kti = 0; kti < NKT; ++kti) {
        const int cur = kti & 1;
        if (kti + 1 < NKT) {
            issue_tile(X, W, m0, n0, (kti + 1) * BK, aB[cur ^ 1], bB[cur ^ 1], tid);
            wait_async_le6();        // current tile's 6 copies retired
        } else {
            wait_async_0();
        }
        __syncthreads();

        const float* A_s = &As[cur][0];
        const float* B_s = &Bs[cur][0];
#pragma unroll
        for (int kk = 0; kk < BK; kk += 4) {
            // A fragment (ISA 16x4 f32): lanes 0-15 -> K=kk,kk+1; 16-31 -> K=kk+2,kk+3
            v2f a[4];
#pragma unroll
            for (int mt = 0; mt < 4; ++mt)
                a[mt] = *(const v2f*)(A_s + (wr * 64 + mt * 16 + lrow) * ASTRIDE
                                          + kk + khalf * 2);
            // B fragment (4x16): K rows striped across lane halves / vgpr slots
            v2f b[4];
#pragma unroll
            for (int nt = 0; nt < 4; ++nt) {
                const int col = wc * 64 + nt * 16 + lrow;
                v2f bb;
                bb.x = B_s[(kk + khalf * 2 + 0) * BSTRIDE + col];
                bb.y = B_s[(kk + khalf * 2 + 1) * BSTRIDE + col];
                b[nt] = bb;
            }
#pragma unroll
            for (int mt = 0; mt < 4; ++mt)
#pragma unroll
                for (int nt = 0; nt < 4; ++nt)
                    acc[mt][nt] = __builtin_amdgcn_wmma_f32_16x16x4_f32(
                        false, a[mt], false, b[nt], (short)0, acc[mt][nt],
                        false, false);
        }
        __syncthreads();
    }

    // C/D layout: VGPR v -> M = base + v + 8*khalf, N = base + (lane&15)
#pragma unroll
    for (int mt = 0; mt < 4; ++mt) {
#pragma unroll
        for (int nt = 0; nt < 4; ++nt) {
            const int mrow = m0 + wr * 64 + mt * 16 + khalf * 8;
            const int ncol = n0 + wc * 64 + nt * 16 + lrow;
#pragma unroll
            for (int v = 0; v < 8; ++v)
                __builtin_nontemporal_store(acc[mt][nt][v],
                    &Z[(size_t)(mrow + v) * N_DIM + ncol]);
        }
    }
}

// Phase 2: in-place SNU recurrence over Z (lives in d_out). One thread per
// (b, n) column, sequential in t; coalesced; non-temporal (streamed once).
__launch_bounds__(256)
__global__ void snu_recur(float* __restrict__ ZY,
                          const float* __restrict__ decay)
{
    const int i = blockIdx.x * 256 + threadIdx.x;   // 0 .. B*N-1
    const float l = decay[i % N_DIM];
    float s = 0.f;
    float reset = 1.f;
#pragma unroll 4
    for (int t = 0; t < T_STEPS; ++t) {
        const size_t off = (size_t)t * (BATCH * N_DIM) + i;
        const float z = __builtin_nontemporal_load(&ZY[off]);
        s = fmaxf(fmaf(l * s, reset, z), 0.f);
        const float y = (s > THRESH) ? 1.f : 0.f;
        __builtin_nontemporal_store(y, &ZY[off]);
        reset = 1.f - y;
    }
}

extern "C" void kernel_launch(void* const* d_in, const int* in_sizes, int n_in,
                              void* d_out, int out_size, void* d_ws, size_t ws_size,
                              hipStream_t stream) {
    const float* x     = (const float*)d_in[0];   // (T, B, IN) fp32
    const float* w_xh  = (const float*)d_in[1];   // (IN, N)    fp32
    const float* decay = (const float*)d_in[2];   // (N,)       fp32
    float* out = (float*)d_out;                   // (T, B, N)  fp32

    dim3 grid(N_DIM / BN, M_DIM / BM);            // (8, 500)
    snu_gemm_f32<<<grid, 256, 0, stream>>>(x, w_xh, out);

    snu_recur<<<(BATCH * N_DIM) / 256, 256, 0, stream>>>(out, decay);
}